// fTNModel_1391569404357
// MI455X (gfx1250) — compile-verified
//
#include <hip/hip_runtime.h>

typedef float v2f __attribute__((ext_vector_type(2)));
typedef float v8f __attribute__((ext_vector_type(8)));

#define LX 6
#define LY 6
#define DDIM 4
#define PP 2
#define NSITE (LX*LY)
#define SITE_ELEMS (DDIM*DDIM*DDIM*DDIM*PP)   // 512
#define NWAVE 8
#define NTHREADS (NWAVE*32)
#define MAXSTATE 16384                         // peak boundary-state size (floats)

struct TileMeta { int w, v, base; bool colok; };

__device__ __forceinline__ TileMeta mk_tile(int n, int l16, int Ncols, int lv, int V,
                                            int HUV, int u0) {
  TileMeta t;
  const int c0 = n * 16 + l16;
  const int c  = (c0 < Ncols) ? c0 : (Ncols - 1);   // clamp; result discarded
  t.colok = (c0 < Ncols);
  t.w = c >> lv;
  t.v = c & (V - 1);
  t.base = t.w * HUV + t.v + u0 * V;
  return t;
}

template <int HC>
__device__ __forceinline__ void load_B(const float* __restrict__ sin, const TileMeta& t,
                                       int UV, int V, v2f (&B)[HC]) {
#pragma unroll
  for (int kc = 0; kc < HC; ++kc) {
    B[kc][0] = sin[t.base + kc * UV];       // rows with u>=U may be garbage:
    B[kc][1] = sin[t.base + kc * UV + V];   //   A is zero there, harmless
  }
}

// One column absorption: out[w,(d,r),v] = sum_{h,u} A[(d,r),(h,u)] * in[w,(h,u),v]
// as 16x16 (A) x 16xN (B) tiles via v_wmma_f32_16x16x4_f32, HC K-chunks each.
// HC/DdT/RT are compile-time edge classes -> branch-free WMMA chain + stores.
template <int HC, int DdT, int RT>
__device__ __forceinline__ void do_column(const float* __restrict__ sin,
                                          float* __restrict__ sout,
                                          const v2f (&A)[4],
                                          int Ncols, int Ntiles, int lv, int V, int UV,
                                          int wave, int lhalf, int l16, int tid) {
  const int u0  = 2 * lhalf;              // B: K_local = g + 2*half
  const int HUV = HC * UV;
  const int DRV = DdT * RT * V;
  // When any padded store exists (DdT/RT==1 or Ncols<16) the output state is
  // <= 4096 floats, so the top NTHREADS slots of sout are a safe dump target.
  float* const dump = &sout[MAXSTATE - NTHREADS + tid];

  int n = wave;
  if (n >= Ntiles) return;
  TileMeta t = mk_tile(n, l16, Ncols, lv, V, HUV, u0);
  v2f Bc[HC];
  load_B<HC>(sin, t, UV, V, Bc);

  while (n < Ntiles) {
    const int n2 = n + NWAVE;
    // Software pipeline: fetch next tile's B before this tile's WMMA chain
    // (last iteration harmlessly re-fetches the current tile).
    TileMeta t2 = mk_tile((n2 < Ntiles) ? n2 : n, l16, Ncols, lv, V, HUV, u0);
    v2f Bn[HC];
    load_B<HC>(sin, t2, UV, V, Bn);

    v8f Cacc = {};
#pragma unroll
    for (int kc = 0; kc < HC; ++kc)
      Cacc = __builtin_amdgcn_wmma_f32_16x16x4_f32(false, A[kc], false, Bc[kc],
                                                   (short)0, Cacc, false, false);

    // C layout: VGPR g holds M = g + 8*(lane>=16), N = lane%16.
    const int wbase = t.w * DRV;
#pragma unroll
    for (int g = 0; g < 8; ++g) {
      const int  M0 = g, M1 = g + 8;                       // M per lane-half
      const bool ok0 = ((M0 >> 2) < DdT) && ((M0 & 3) < RT);  // compile-time
      const bool ok1 = ((M1 >> 2) < DdT) && ((M1 & 3) < RT);  // compile-time
      if (ok0 || ok1) {                                    // else store elided
        const int idx0 = (M0 >> 2) * RT + (M0 & 3);        // compile-time
        const int idx1 = (M1 >> 2) * RT + (M1 & 3);        // compile-time
        const bool ok  = (lhalf ? ok1 : ok0) && t.colok;
        const int  idx = wbase + (lhalf ? idx1 : idx0) * V + t.v;
        float* dst = ok ? &sout[idx] : dump;
        *dst = Cacc[g];
      }
    }

    t = t2;
#pragma unroll
    for (int kc = 0; kc < HC; ++kc) Bc[kc] = Bn[kc];       // renamed away by RA
    n = n2;
  }
}

// One workgroup (8 wave32) per batch item; boundary state ping-pongs in LDS.
__global__ __launch_bounds__(NTHREADS) void tn_amp_kernel(const int* __restrict__ x,
                                                          const float* __restrict__ tensors,
                                                          float* __restrict__ out) {
  __shared__ float S0[MAXSTATE];
  __shared__ float S1[MAXSTATE];

  const int b     = blockIdx.x;
  const int tid   = threadIdx.x;
  const int wave  = tid >> 5;
  const int lane  = tid & 31;
  const int lhalf = lane >> 4;
  const int l16   = lane & 15;

  if (tid == 0) S0[0] = 1.0f;          // initial boundary vector = [1]
  __syncthreads();

  float* sin  = S0;
  float* sout = S1;
  int state = 1;                       // current boundary-state element count

  for (int i = 0; i < LX; ++i) {
    int W = 1;                                   // processed-columns bond product
    const int U  = (i == 0)      ? 1 : 4;        // up-bond size this row
    const int Dd = (i == LX - 1) ? 1 : 4;        // down-bond size this row
    const bool lastrow = (i == LX - 1);
    for (int j = 0; j < LY; ++j) {
      const int Hc = (j == 0)      ? 1 : 4;      // left (horizontal-in) bond
      const int R  = (j == LY - 1) ? 1 : 4;      // right (horizontal-out) bond
      const int V  = state / (W * Hc * U);       // trailing un-absorbed bonds
      const int lv = 31 - __clz(V);              // V is a power of two
      const int Ncols  = W * V;
      const int Ntiles = (Ncols + 15) >> 4;
      const int UV = U * V;

      // Site tensor, physical index p = x[b,i,j] (uniform scalar load).
      const int p = x[b * NSITE + i * LY + j];
      const float* T = tensors + (i * LY + j) * SITE_ELEMS + p;

      // A[(d,r),(h,u)] = T[u,d,h,r], zero-padded to 16x16, 4 K-chunks (16x4).
      // ISA 32-bit A layout: M = lane%16; K_local = g + 2*(lane>=16).
      // Gather indices always in-bounds -> unconditional load + cndmask.
      const int d_a = l16 >> 2;
      const int r_a = l16 & 3;
      v2f A[4];
#pragma unroll
      for (int kc = 0; kc < 4; ++kc) {
#pragma unroll
        for (int g = 0; g < 2; ++g) {
          const int u_a = g + 2 * lhalf;
          const int h_a = kc;
          const float av = T[(((u_a * DDIM + d_a) * DDIM + h_a) * DDIM + r_a) * PP];
          const bool ok = (u_a < U) & (h_a < Hc) & (d_a < Dd) & (r_a < R);
          A[kc][g] = ok ? av : 0.0f;
        }
      }

      // 6-way compile-time edge-class dispatch (branch-free inner loops).
      if (!lastrow) {
        if (j == 0)           do_column<1, 4, 4>(sin, sout, A, Ncols, Ntiles, lv, V, UV, wave, lhalf, l16, tid);
        else if (j < LY - 1)  do_column<4, 4, 4>(sin, sout, A, Ncols, Ntiles, lv, V, UV, wave, lhalf, l16, tid);
        else                  do_column<4, 4, 1>(sin, sout, A, Ncols, Ntiles, lv, V, UV, wave, lhalf, l16, tid);
      } else {
        if (j == 0)           do_column<1, 1, 4>(sin, sout, A, Ncols, Ntiles, lv, V, UV, wave, lhalf, l16, tid);
        else if (j < LY - 1)  do_column<4, 1, 4>(sin, sout, A, Ncols, Ntiles, lv, V, UV, wave, lhalf, l16, tid);
        else                  do_column<4, 1, 1>(sin, sout, A, Ncols, Ntiles, lv, V, UV, wave, lhalf, l16, tid);
      }

      __syncthreads();
      float* t = sin; sin = sout; sout = t;
      state = W * Dd * R * V;
      W *= Dd;
    }
  }
  if (tid == 0) out[b] = sin[0];       // final 1-element boundary = amplitude
}

extern "C" void kernel_launch(void* const* d_in, const int* in_sizes, int n_in,
                              void* d_out, int out_size, void* d_ws, size_t ws_size,
                              hipStream_t stream) {
  const int*   xin     = (const int*)d_in[0];      // (1024, 36) int32
  const float* tensors = (const float*)d_in[1];    // (6,6,4,4,4,4,2) f32
  float*       outp    = (float*)d_out;            // (1024,) f32
  (void)in_sizes; (void)n_in; (void)d_ws; (void)ws_size;
  tn_amp_kernel<<<dim3(out_size), dim3(NTHREADS), 0, stream>>>(xin, tensors, outp);
}